// TrainNet_11922829214311
// MI455X (gfx1250) — compile-verified
//
#include <hip/hip_runtime.h>

#define OUT_F 4096
#define IN_F  4096
#define NCOLS 8192
#define KB    32
#define AS_STRIDE 36    // 32 + pad; 36*lo mod 64 distinct for lo=0..15, float4-aligned
#define BS_STRIDE 132   // 128 + pad; rows k and k+2 land on different banks

typedef float v2f __attribute__((ext_vector_type(2)));
typedef float v8f __attribute__((ext_vector_type(8)));

// ---- Phase 0: zero the dense-W workspace (graph-capture-safe) ----
__global__ __launch_bounds__(256) void zero_ws(float4* __restrict__ p, int n4) {
    int i = blockIdx.x * 256 + threadIdx.x;
    if (i < n4) p[i] = make_float4(0.f, 0.f, 0.f, 0.f);
}

// ---- Phase 1: COO scatter into dense W; duplicate (row,col) entries sum ----
__global__ __launch_bounds__(256) void coo_scatter(const int* __restrict__ rows,
                                                   const int* __restrict__ cols,
                                                   const float* __restrict__ vals,
                                                   float* __restrict__ W, int nnz) {
    int i = blockIdx.x * 256 + threadIdx.x;
    if (i < nnz) atomicAdd(&W[rows[i] * IN_F + cols[i]], vals[i]);
}

// ---- Phase 2: dense fp32 WMMA GEMM: out = W(4096x4096) @ x(4096x8192) ----
// Block tile 64(M) x 128(N), 8 wave32s in a 2x4 grid, wave tile 32x32.
// Double-buffered LDS + register-staged global loads: one barrier per K=32
// chunk, next chunk's HBM/L2 loads overlap the current chunk's 32 WMMAs.
__global__ __launch_bounds__(256) void wmma_gemm(const float* __restrict__ W,
                                                 const float* __restrict__ x,
                                                 float* __restrict__ out) {
    __shared__ float As[2][64 * AS_STRIDE];   // 64 rows x 32 K (padded), x2 buffers
    __shared__ float Bs[2][KB * BS_STRIDE];   // 32 K x 128 cols (padded), x2 buffers

    const int tid  = threadIdx.x;
    const int wid  = tid >> 5;
    const int lane = tid & 31;
    const int half = lane >> 4;   // 0: lanes 0-15, 1: lanes 16-31
    const int lo   = lane & 15;

    const int mb0 = blockIdx.y * 64;
    const int nb0 = blockIdx.x * 128;
    const int ma  = (wid >> 2) * 32;   // wave row offset in block tile
    const int na  = (wid & 3) * 32;    // wave col offset in block tile

    v8f c00 = {}; v8f c01 = {}; v8f c10 = {}; v8f c11 = {};

    // A staging: row = tid>>2 (0..63), cols (tid&3)*8 + {0,4}  -> 2 float4 / thread
    const int ar = tid >> 2;
    const int ac = (tid & 3) << 3;
    const float* Aptr = W + (mb0 + ar) * IN_F + ac;
    // B staging: rows br+{0,8,16,24} (0..31), colv (tid&31)*4 -> 4 float4 / thread
    const int br = tid >> 5;
    const int bc = (tid & 31) << 2;
    const float* Bptr = x + (size_t)br * NCOLS + nb0 + bc;

    float4 a0r, a1r, b0r, b1r, b2r, b3r;

    // Prologue: fetch chunk 0 into registers.
    a0r = *(const float4*)(Aptr + 0);
    a1r = *(const float4*)(Aptr + 4);
    b0r = *(const float4*)(Bptr + (size_t)0 * 8 * NCOLS);
    b1r = *(const float4*)(Bptr + (size_t)1 * 8 * NCOLS);
    b2r = *(const float4*)(Bptr + (size_t)2 * 8 * NCOLS);
    b3r = *(const float4*)(Bptr + (size_t)3 * 8 * NCOLS);

    for (int kb = 0; kb < IN_F; kb += KB) {
        float* AsB = As[(kb >> 5) & 1];
        float* BsB = Bs[(kb >> 5) & 1];

        // Commit the register-staged chunk to this iteration's LDS buffer.
        *(float4*)&AsB[ar * AS_STRIDE + ac]        = a0r;
        *(float4*)&AsB[ar * AS_STRIDE + ac + 4]    = a1r;
        *(float4*)&BsB[(br +  0) * BS_STRIDE + bc] = b0r;
        *(float4*)&BsB[(br +  8) * BS_STRIDE + bc] = b1r;
        *(float4*)&BsB[(br + 16) * BS_STRIDE + bc] = b2r;
        *(float4*)&BsB[(br + 24) * BS_STRIDE + bc] = b3r;
        __syncthreads();   // single barrier per chunk (double buffering covers WAR)

        // Issue next chunk's global loads; latency hides behind 32 WMMAs below.
        if (kb + KB < IN_F) {
            const float* Ap = Aptr + (kb + KB);
            const float* Bp = Bptr + (size_t)(kb + KB) * NCOLS;
            a0r = *(const float4*)(Ap + 0);
            a1r = *(const float4*)(Ap + 4);
            b0r = *(const float4*)(Bp + (size_t)0 * 8 * NCOLS);
            b1r = *(const float4*)(Bp + (size_t)1 * 8 * NCOLS);
            b2r = *(const float4*)(Bp + (size_t)2 * 8 * NCOLS);
            b3r = *(const float4*)(Bp + (size_t)3 * 8 * NCOLS);
        }

#pragma unroll
        for (int k0 = 0; k0 < KB; k0 += 4) {
            const int kr = k0 + 2 * half;   // per-lane K base (ISA A/B fragment layout)
            v2f a0 = *(const v2f*)&AsB[(ma + lo) * AS_STRIDE + kr];
            v2f a1 = *(const v2f*)&AsB[(ma + 16 + lo) * AS_STRIDE + kr];
            v2f b0, b1;
            b0.x = BsB[kr * BS_STRIDE + na + lo];
            b0.y = BsB[(kr + 1) * BS_STRIDE + na + lo];
            b1.x = BsB[kr * BS_STRIDE + na + 16 + lo];
            b1.y = BsB[(kr + 1) * BS_STRIDE + na + 16 + lo];
            // 4 independent accumulation chains hide XDL latency.
            c00 = __builtin_amdgcn_wmma_f32_16x16x4_f32(false, a0, false, b0, (short)0, c00, false, false);
            c01 = __builtin_amdgcn_wmma_f32_16x16x4_f32(false, a0, false, b1, (short)0, c01, false, false);
            c10 = __builtin_amdgcn_wmma_f32_16x16x4_f32(false, a1, false, b0, (short)0, c10, false, false);
            c11 = __builtin_amdgcn_wmma_f32_16x16x4_f32(false, a1, false, b1, (short)0, c11, false, false);
        }
        // No trailing barrier: the next iteration writes the *other* buffer,
        // and its collective barrier already ordered the previous readers.
    }

    // C/D layout: VGPR j holds (M = j + 8*half, N = lo).
    const int row0 = mb0 + ma + 8 * half;
    const int col0 = nb0 + na + lo;
#pragma unroll
    for (int j = 0; j < 8; ++j) {
        out[(row0 + j) * NCOLS + col0]           = c00[j];
        out[(row0 + j) * NCOLS + col0 + 16]      = c01[j];
        out[(row0 + 16 + j) * NCOLS + col0]      = c10[j];
        out[(row0 + 16 + j) * NCOLS + col0 + 16] = c11[j];
    }
}

extern "C" void kernel_launch(void* const* d_in, const int* in_sizes, int n_in,
                              void* d_out, int out_size, void* d_ws, size_t ws_size,
                              hipStream_t stream) {
    const int*   rows = (const int*)d_in[0];
    const int*   cols = (const int*)d_in[1];
    const float* vals = (const float*)d_in[2];
    const float* x    = (const float*)d_in[3];
    float*       out  = (float*)d_out;
    float*       W    = (float*)d_ws;   // 4096*4096 fp32 = 64 MB dense weight
    const int    nnz  = in_sizes[0];

    const int n4 = OUT_F * IN_F / 4;
    zero_ws<<<(n4 + 255) / 256, 256, 0, stream>>>((float4*)W, n4);
    coo_scatter<<<(nnz + 255) / 256, 256, 0, stream>>>(rows, cols, vals, W, nnz);
    wmma_gemm<<<dim3(NCOLS / 128, OUT_F / 64), 256, 0, stream>>>(W, x, out);
}